// QuantumLayer_16355235463230
// MI455X (gfx1250) — compile-verified
//
#include <hip/hip_runtime.h>
#include <math.h>

#define NQ      14
#define NL      4
#define BATCHN  128
#define DIM     16384   // 2^14 amplitudes
#define THREADS 256     // 8 wave32

typedef float v2f __attribute__((ext_vector_type(2)));
typedef float v8f __attribute__((ext_vector_type(8)));

struct C2x2 { float r[2][2]; float i[2][2]; };

// PennyLane Rot(phi,theta,omega) = RZ(omega) RY(theta) RZ(phi)
__device__ inline void rotMat(float phi, float theta, float omega, C2x2& M) {
  float ct = cosf(0.5f * theta), st = sinf(0.5f * theta);
  float apo = 0.5f * (phi + omega);
  float amo = 0.5f * (phi - omega);
  float c1 = cosf(apo), s1 = sinf(apo);
  float c2 = cosf(amo), s2 = sinf(amo);
  M.r[0][0] =  c1 * ct;  M.i[0][0] = -s1 * ct;   // e^{-i(phi+omega)/2} c
  M.r[0][1] = -c2 * st;  M.i[0][1] = -s2 * st;   // -e^{+i(phi-omega)/2} s
  M.r[1][0] =  c2 * st;  M.i[1][0] = -s2 * st;   // e^{-i(phi-omega)/2} s
  M.r[1][1] =  c1 * ct;  M.i[1][1] =  s1 * ct;   // e^{+i(phi+omega)/2} c
}

// Generic single-qubit gate: 8192 disjoint amplitude pairs, 32 per thread.
__device__ void apply1q(float* sRe, float* sIm, int wire, const C2x2& M) {
  const int s = 1 << (13 - wire);
  for (int p = threadIdx.x; p < DIM / 2; p += THREADS) {
    int lo = p & (s - 1);
    int i0 = ((p & ~(s - 1)) << 1) | lo;
    int i1 = i0 + s;
    float a0r = sRe[i0], a0i = sIm[i0];
    float a1r = sRe[i1], a1i = sIm[i1];
    float n0r = M.r[0][0]*a0r - M.i[0][0]*a0i + M.r[0][1]*a1r - M.i[0][1]*a1i;
    float n0i = M.r[0][0]*a0i + M.i[0][0]*a0r + M.r[0][1]*a1i + M.i[0][1]*a1r;
    float n1r = M.r[1][0]*a0r - M.i[1][0]*a0i + M.r[1][1]*a1r - M.i[1][1]*a1i;
    float n1i = M.r[1][0]*a0i + M.i[1][0]*a0r + M.r[1][1]*a1i + M.i[1][1]*a1r;
    sRe[i0] = n0r; sIm[i0] = n0i;
    sRe[i1] = n1r; sIm[i1] = n1i;
  }
  __syncthreads();
}

// CNOT(c,t): swap amplitudes with bit_c=1 between target=0/1. 4096 disjoint swaps.
__device__ void applyCnot(float* sRe, float* sIm, int c, int t) {
  const int pc = 13 - c, pt = 13 - t;
  const int pl = pc < pt ? pc : pt;
  const int ph = pc < pt ? pt : pc;
  const int ml = 1 << pl, mh = 1 << ph;
  for (int q = threadIdx.x; q < DIM / 4; q += THREADS) {
    int i = q;
    i = ((i & ~(ml - 1)) << 1) | (i & (ml - 1));   // insert 0 at low pos
    i = ((i & ~(mh - 1)) << 1) | (i & (mh - 1));   // insert 0 at high pos
    i |= (1 << pc);                                 // control bit = 1, target bit = 0
    int j = i | (1 << pt);
    float tr = sRe[i]; sRe[i] = sRe[j]; sRe[j] = tr;
    float ti = sIm[i]; sIm[i] = sIm[j]; sIm[j] = ti;
  }
  __syncthreads();
}

// U = M0 (x) M1 (x) M2 (x) M3, 16x16 complex, one element per thread.
__device__ void buildKron(float* uRe, float* uIm, const C2x2 M[4]) {
  const int t = threadIdx.x;
  const int a = t >> 4, b = t & 15;
  float pr = 1.f, pi = 0.f;
#pragma unroll
  for (int q = 0; q < 4; ++q) {
    int ab = (a >> (3 - q)) & 1;
    int bb = (b >> (3 - q)) & 1;
    float mr = M[q].r[ab][bb], mi = M[q].i[ab][bb];
    float nr = pr * mr - pi * mi;
    float ni = pr * mi + pi * mr;
    pr = nr; pi = ni;
  }
  uRe[a * 16 + b] = pr;
  uIm[a * 16 + b] = pi;
  __syncthreads();
}

__device__ inline v8f wmma_f32(v2f a, v2f b, v8f c) {
  // 8 args: (neg_a, A, neg_b, B, c_mod, C, reuse_a, reuse_b)
  return __builtin_amdgcn_wmma_f32_16x16x4_f32(false, a, false, b, (short)0, c,
                                               false, false);
}

// Apply fused 16x16 unitary on 4 qubits whose index bits sit at position p..p+3.
// new[m, n] = sum_g U[n][g] * state[m, g], m over 1024 (hi,lo) rows.
// 64 row-tiles of 16; 8 waves x 8 tiles; rows are tile-exclusive -> no inner barrier.
__device__ void applyGroupWMMA(float* sRe, float* sIm,
                               const float* uRe, const float* uIm, int p) {
  const int lane = threadIdx.x & 31;
  const int wave = threadIdx.x >> 5;
  const int n    = lane & 15;
  const int half = lane >> 4;          // 0 or 1

  // B fragments: B[k][n] = U[n][k]; layout: VGPR j, lane-half h -> K = j + 2h.
  v2f bre[4], bim[4];
#pragma unroll
  for (int ks = 0; ks < 4; ++ks) {
    int k = ks * 4 + half * 2;
    bre[ks][0] = uRe[n * 16 + k];     bre[ks][1] = uRe[n * 16 + k + 1];
    bim[ks][0] = uIm[n * 16 + k];     bim[ks][1] = uIm[n * 16 + k + 1];
  }

  const int pm = (1 << p) - 1;
  for (int tile = wave; tile < 64; tile += 8) {
    // A layout: lanes 0-15 hold M=L (K pair 0,1), lanes 16-31 hold M=L-16 (K pair 2,3)
    int mrow = tile * 16 + n;
    int base = ((mrow & ~pm) << 4) | (mrow & pm);   // hi*2^(p+4) + lo
    v8f cre = {0.f,0.f,0.f,0.f,0.f,0.f,0.f,0.f};
    v8f cim = {0.f,0.f,0.f,0.f,0.f,0.f,0.f,0.f};
#pragma unroll
    for (int ks = 0; ks < 4; ++ks) {
      int k  = ks * 4 + half * 2;
      int o0 = base + (k << p);
      int o1 = base + ((k + 1) << p);
      v2f are, aim, nim;
      are[0] = sRe[o0]; are[1] = sRe[o1];
      aim[0] = sIm[o0]; aim[1] = sIm[o1];
      nim[0] = -aim[0]; nim[1] = -aim[1];
      // complex: Cre += Ar*Br - Ai*Bi ; Cim += Ar*Bi + Ai*Br
      cre = wmma_f32(are, bre[ks], cre);
      cre = wmma_f32(nim, bim[ks], cre);
      cim = wmma_f32(are, bim[ks], cim);
      cim = wmma_f32(aim, bre[ks], cim);
    }
    // D layout: VGPR v, lane-half h -> M = v + 8h, N = lane%16
#pragma unroll
    for (int v = 0; v < 8; ++v) {
      int m   = tile * 16 + v + half * 8;
      int bse = ((m & ~pm) << 4) | (m & pm);
      int idx = bse + (n << p);
      sRe[idx] = cre[v];
      sIm[idx] = cim[v];
    }
  }
  __syncthreads();
}

__global__ __launch_bounds__(THREADS)
void qsim_kernel(const float* __restrict__ x, const float* __restrict__ W,
                 float* __restrict__ out) {
  __shared__ float sRe[DIM];
  __shared__ float sIm[DIM];
  __shared__ float uRe[256];
  __shared__ float uIm[256];
  __shared__ float redBuf[THREADS * NQ];
  __shared__ int oob;

  const int b   = blockIdx.x;
  const int tid = threadIdx.x;

  if (tid == 0) oob = 0;
  for (int i = tid; i < DIM; i += THREADS) { sRe[i] = 0.f; sIm[i] = 0.f; }
  __syncthreads();

  // jnp.all over the ENTIRE x tensor decides atan remap for everyone.
  int loc = 0;
  for (int i = tid; i < BATCHN * NQ; i += THREADS) {
    float v = x[i];
    if (!(v >= -3.15f && v <= 3.15f)) loc = 1;
  }
  if (loc) oob = 1;                 // benign race: all writers store 1
  if (tid == 0) sRe[0] = 1.f;       // |0...0>
  __syncthreads();
  const bool useAtan = (oob != 0);

  // AngleEmbedding: RX(x[b,w]) on each wire
  for (int w = 0; w < NQ; ++w) {
    float xv = x[b * NQ + w];
    if (useAtan) xv = atanf(xv);
    float c = cosf(0.5f * xv), s = sinf(0.5f * xv);
    C2x2 M;
    M.r[0][0] = c;   M.i[0][0] = 0.f;
    M.r[0][1] = 0.f; M.i[0][1] = -s;
    M.r[1][0] = 0.f; M.i[1][0] = -s;
    M.r[1][1] = c;   M.i[1][1] = 0.f;
    apply1q(sRe, sIm, w, M);
  }

  // StronglyEntanglingLayers
  const int w0s[3] = {2, 6, 10};
  const int ps[3]  = {8, 4, 0};
  for (int l = 0; l < NL; ++l) {
    // wires 0,1: plain pair-sweeps (index bits 13,12 -> 4x4 group not worth WMMA)
    for (int w = 0; w < 2; ++w) {
      C2x2 M;
      rotMat(W[(l*NQ+w)*3+0], W[(l*NQ+w)*3+1], W[(l*NQ+w)*3+2], M);
      apply1q(sRe, sIm, w, M);
    }
    // wires 2-13: three fused 4-qubit groups via WMMA
    for (int g = 0; g < 3; ++g) {
      C2x2 M[4];
#pragma unroll
      for (int q = 0; q < 4; ++q) {
        int w = w0s[g] + q;
        rotMat(W[(l*NQ+w)*3+0], W[(l*NQ+w)*3+1], W[(l*NQ+w)*3+2], M[q]);
      }
      buildKron(uRe, uIm, M);
      applyGroupWMMA(sRe, sIm, uRe, uIm, ps[g]);
    }
    // CNOT ring, range r, applied sequentially (they don't all commute)
    int r = (l % (NQ - 1)) + 1;
    for (int w = 0; w < NQ; ++w)
      applyCnot(sRe, sIm, w, (w + r) % NQ);
  }

  // <Z_w> = sum_i p_i * (1 - 2*bit_w(i))
  float acc[NQ];
#pragma unroll
  for (int w = 0; w < NQ; ++w) acc[w] = 0.f;
  for (int i = tid; i < DIM; i += THREADS) {
    float pr = sRe[i]*sRe[i] + sIm[i]*sIm[i];
#pragma unroll
    for (int w = 0; w < NQ; ++w)
      acc[w] += ((i >> (13 - w)) & 1) ? -pr : pr;
  }
#pragma unroll
  for (int w = 0; w < NQ; ++w) redBuf[tid * NQ + w] = acc[w];
  __syncthreads();
  if (tid < NQ) {
    float ssum = 0.f;
    for (int t2 = 0; t2 < THREADS; ++t2) ssum += redBuf[t2 * NQ + tid];
    out[b * NQ + tid] = ssum;
  }
}

extern "C" void kernel_launch(void* const* d_in, const int* in_sizes, int n_in,
                              void* d_out, int out_size, void* d_ws, size_t ws_size,
                              hipStream_t stream) {
  (void)in_sizes; (void)n_in; (void)d_ws; (void)ws_size; (void)out_size;
  const float* x = (const float*)d_in[0];     // (128, 14) f32
  const float* W = (const float*)d_in[1];     // (4, 14, 3) f32
  float* out = (float*)d_out;                 // (128, 14) f32
  qsim_kernel<<<BATCHN, THREADS, 0, stream>>>(x, W, out);
}